// Model_70557722739215
// MI455X (gfx1250) — compile-verified
//
#include <hip/hip_runtime.h>
#include <hip/hip_bf16.h>
#include <cstddef>

// ---------------------------------------------------------------------------
// ST-GCN block on MI455X (gfx1250, wave32).
//
//   N=4, C_IN=C_OUT=64, L=1024, V=25, P=3, window=9 (stride 1), EPS=1e-5
//
// Restructured as:
//   Mc[(ci,v),(c,w)] = sum_p conv_w[p*64+c,ci] * A[p,v,w] * edge[p,v,w]
//   bias_cw[c,w]     = sum_p conv_b[p*64+c] * sum_v A[p,v,w]*edge[p,v,w]
//   Y[(n,l),(c,w)]   = X[(n,l),(ci,v)] @ Mc + bias_cw      (4096x1600x1600 GEMM)
//   y4[n,c,l,w]      = sum_{j=l-8..l} Y[(n,j),(c,w)]       (sliding window)
//   out              = relu(relu(BN_c(y4)) + x)
//
// GEMM uses V_WMMA_F32_16X16X4_F32 (exact fp32 matrix path), 2x2 register
// tile per wave for fragment reuse (2 LDS floats per WMMA).
// ---------------------------------------------------------------------------

typedef float v2f __attribute__((ext_vector_type(2)));
typedef float v8f __attribute__((ext_vector_type(8)));

#define NN    4
#define CC    64      // C_IN == C_OUT
#define LLEN  1024
#define VV    25
#define PPART 3
#define KDIM  1600    // C_IN * V
#define NDIM  1600    // C_OUT * V
#define MDIM  4096    // N * L
#define NELEM 6553600 // N*C*L*V

// workspace layout (in floats)
#define OFF_MC    ((size_t)0)                 // 1600*1600   = 2,560,000
#define OFF_XP    ((size_t)2560000)           // 4096*1600   = 6,553,600
#define OFF_Y     ((size_t)(2560000+6553600))           // 6,553,600
#define OFF_Y4    ((size_t)(2560000+2*6553600))         // 6,553,600
#define OFF_BIAS  ((size_t)(2560000+3*6553600))         // 1,600
#define OFF_STATS ((size_t)(2560000+3*6553600+1600))    // 256
// total = 22,222,656 floats = ~88.9 MB

// --------------------------------------------------------------------------
// Kernel 1: combined matrix Mc[(ci*25+v)][(c*25+w)]
// --------------------------------------------------------------------------
__global__ void build_mc(const float* __restrict__ cw,   // (P*64, 64)
                         const float* __restrict__ A,    // (P,25,25)
                         const float* __restrict__ ei,   // (P,25,25)
                         float* __restrict__ mc) {
    int id = blockIdx.x * 256 + threadIdx.x;
    if (id >= KDIM * NDIM) return;
    int row = id / NDIM;
    int col = id - row * NDIM;
    int ci = row / VV, v = row - ci * VV;
    int c  = col / VV, w = col - c * VV;
    float acc = 0.f;
#pragma unroll
    for (int p = 0; p < PPART; ++p) {
        float aw = A[(p * VV + v) * VV + w] * ei[(p * VV + v) * VV + w];
        acc += cw[(p * CC + c) * CC + ci] * aw;
    }
    mc[id] = acc;
}

// --------------------------------------------------------------------------
// Kernel 2: bias_cw + zero stats accumulators
// --------------------------------------------------------------------------
__global__ void build_bias(const float* __restrict__ cb,  // (P*64)
                           const float* __restrict__ A,
                           const float* __restrict__ ei,
                           float* __restrict__ bias,
                           float* __restrict__ stats) {
    int id = blockIdx.x * 256 + threadIdx.x;
    if (id < 128) stats[id] = 0.f;   // [0..63]=sum, [64..127]=sumsq
    if (id >= NDIM) return;
    int c = id / VV, w = id - c * VV;
    float s = 0.f;
#pragma unroll
    for (int p = 0; p < PPART; ++p) {
        float sv = 0.f;
        for (int v = 0; v < VV; ++v)
            sv += A[(p * VV + v) * VV + w] * ei[(p * VV + v) * VV + w];
        s += cb[p * CC + c] * sv;
    }
    bias[id] = s;
}

// --------------------------------------------------------------------------
// Kernel 3: pack x(n,ci,l,v) -> Xp[(n*L+l)][(ci*25+v)]  (GEMM A operand)
// --------------------------------------------------------------------------
__global__ void pack_x(const float* __restrict__ x, float* __restrict__ xp) {
    int id = blockIdx.x * 256 + threadIdx.x;
    if (id >= MDIM * KDIM) return;
    int row = id / KDIM;
    int k   = id - row * KDIM;
    int n = row >> 10, l = row & 1023;
    int ci = k / VV, v = k - ci * VV;
    xp[id] = x[(((size_t)n * CC + ci) * LLEN + l) * VV + v];
}

// --------------------------------------------------------------------------
// Kernel 4: GEMM  Y = Xp @ Mc + bias  using V_WMMA_F32_16X16X4_F32.
//   block: 256 threads = 8 wave32, macro tile 128(M) x 64(N).
//   waves in 4x2 grid; each wave: 32x32 (2x2 fragments of 16x16).
//   Both LDS tiles stored k-fastest with pad-17 rows so each fragment's two
//   k-elements are adjacent (single ds_load_2addr) and reads are
//   conflict-free across the full wave.
// A fragment (16x4 f32): lane = m(0..15) | half; vgpr j -> K = half*2 + j
// B fragment (4x16 f32): lane = n(0..15) | half; vgpr j -> K = half*2 + j
// C/D (16x16 f32): vgpr r -> M = r + half*8, N = lane&15
// --------------------------------------------------------------------------
__global__ __launch_bounds__(256) void gemm_wmma(
        const float* __restrict__ Xp, const float* __restrict__ Mc,
        const float* __restrict__ bias, float* __restrict__ Y) {
    __shared__ float As[128 * 17];  // [m][k], pad 17
    __shared__ float Bs[64 * 17];   // [n][k], pad 17 (B tile transposed)

    const int tid  = threadIdx.x;
    const int lane = tid & 31;
    const int wave = tid >> 5;
    const int wm   = wave >> 1;        // 0..3 : 32-row stripe
    const int wn   = wave & 1;         // 0..1 : 32-col stripe
    const int half = lane >> 4;        // 0/1
    const int idx  = lane & 15;        // 0..15
    const int m0 = blockIdx.x * 128;
    const int n0 = blockIdx.y * 64;

    // cooperative load indices
    const int ar = tid >> 2;           // 0..63 (+64 for second row)
    const int ak = (tid & 3) * 4;      // 0,4,8,12
    const int bk = tid >> 4;           // 0..15  (k row in Mc tile)
    const int bc = (tid & 15) * 4;     // 0..60  (n col in Mc tile)

    v8f acc[2][2] = {};

    for (int k0 = 0; k0 < KDIM; k0 += 16) {
        const float4 a0 = *(const float4*)(Xp + (size_t)(m0 + ar) * KDIM + k0 + ak);
        const float4 a1 = *(const float4*)(Xp + (size_t)(m0 + 64 + ar) * KDIM + k0 + ak);
        const float4 bv = *(const float4*)(Mc + (size_t)(k0 + bk) * NDIM + n0 + bc);
        As[ar * 17 + ak + 0] = a0.x;
        As[ar * 17 + ak + 1] = a0.y;
        As[ar * 17 + ak + 2] = a0.z;
        As[ar * 17 + ak + 3] = a0.w;
        As[(64 + ar) * 17 + ak + 0] = a1.x;
        As[(64 + ar) * 17 + ak + 1] = a1.y;
        As[(64 + ar) * 17 + ak + 2] = a1.z;
        As[(64 + ar) * 17 + ak + 3] = a1.w;
        Bs[(bc + 0) * 17 + bk] = bv.x;   // transpose into [n][k]
        Bs[(bc + 1) * 17 + bk] = bv.y;
        Bs[(bc + 2) * 17 + bk] = bv.z;
        Bs[(bc + 3) * 17 + bk] = bv.w;
        if (k0 + 16 < KDIM) {   // prefetch next K tile -> global_prefetch_b8
            __builtin_prefetch(Xp + (size_t)(m0 + ar) * KDIM + k0 + 16 + ak, 0, 3);
            __builtin_prefetch(Xp + (size_t)(m0 + 64 + ar) * KDIM + k0 + 16 + ak, 0, 3);
            __builtin_prefetch(Mc + (size_t)(k0 + 16 + bk) * NDIM + n0 + bc, 0, 3);
        }
        __syncthreads();

#pragma unroll
        for (int ks = 0; ks < 4; ++ks) {
            const int kb = ks * 4 + half * 2;
            v2f a[2], b[2];
#pragma unroll
            for (int am = 0; am < 2; ++am) {
                const int m = wm * 32 + am * 16 + idx;
                a[am].x = As[m * 17 + kb];
                a[am].y = As[m * 17 + kb + 1];
            }
#pragma unroll
            for (int bn = 0; bn < 2; ++bn) {
                const int n = wn * 32 + bn * 16 + idx;
                b[bn].x = Bs[n * 17 + kb];
                b[bn].y = Bs[n * 17 + kb + 1];
            }
#pragma unroll
            for (int am = 0; am < 2; ++am)
#pragma unroll
                for (int bn = 0; bn < 2; ++bn)
                    acc[am][bn] = __builtin_amdgcn_wmma_f32_16x16x4_f32(
                        false, a[am], false, b[bn], (short)0, acc[am][bn],
                        false, false);
        }
        __syncthreads();
    }

#pragma unroll
    for (int am = 0; am < 2; ++am) {
#pragma unroll
        for (int bn = 0; bn < 2; ++bn) {
            const int ncol = n0 + wn * 32 + bn * 16 + idx;
            const float bval = bias[ncol];
#pragma unroll
            for (int r = 0; r < 8; ++r) {
                const int m = wm * 32 + am * 16 + half * 8 + r;
                Y[(size_t)(m0 + m) * NDIM + ncol] = acc[am][bn][r] + bval;
            }
        }
    }
}

// --------------------------------------------------------------------------
// Kernel 5: width-9 sliding-window sum along l (+ per-channel stat partials)
//   grid (10, N, 8): blockDim 160; thread owns one (c,w), walks 128 l's.
//   y4 stored as (n,c,l,w) so the epilogue is fully coalesced.
// --------------------------------------------------------------------------
__global__ void window_sum(const float* __restrict__ Y,
                           float* __restrict__ y4,
                           float* __restrict__ stats) {
    const int cw = blockIdx.x * 160 + threadIdx.x;       // 0..1599
    const int n  = blockIdx.y;
    const int l0 = blockIdx.z * 128;
    const int c  = cw / VV, w = cw - c * VV;

    const float* Yb = Y + (size_t)n * LLEN * NDIM + cw;
    float* y4b = y4 + ((size_t)(n * CC + c) * LLEN) * VV + w;

    float s = 0.f;
    for (int j = l0 - 9; j < l0; ++j)
        if (j >= 0) s += Yb[(size_t)j * NDIM];

    float ls = 0.f, lss = 0.f;
    for (int l = l0; l < l0 + 128; ++l) {
        s += Yb[(size_t)l * NDIM];
        if (l >= 9) s -= Yb[(size_t)(l - 9) * NDIM];
        y4b[(size_t)l * VV] = s;
        ls += s;
        lss += s * s;
    }
    atomicAdd(&stats[c], ls);
    atomicAdd(&stats[64 + c], lss);
}

// --------------------------------------------------------------------------
// Kernel 6: finalize BN stats -> scale/shift per channel
// --------------------------------------------------------------------------
__global__ void finalize_stats(float* __restrict__ stats,
                               const float* __restrict__ gamma,
                               const float* __restrict__ beta) {
    int c = threadIdx.x;
    const float cnt = (float)(NN * LLEN * VV);           // 102400
    float mean = stats[c] / cnt;
    float var  = stats[64 + c] / cnt - mean * mean;
    float inv  = rsqrtf(var + 1e-5f);
    float sc   = inv * gamma[c];
    stats[128 + c] = sc;
    stats[192 + c] = beta[c] - mean * sc;
}

// --------------------------------------------------------------------------
// Kernel 7: BN + relu + residual + relu, fully coalesced over (n,c,l,v)
// --------------------------------------------------------------------------
__global__ void bn_epilogue(const float* __restrict__ y4,
                            const float* __restrict__ stats,
                            const float* __restrict__ x,
                            float* __restrict__ out) {
    int id = blockIdx.x * 256 + threadIdx.x;
    if (id >= NELEM) return;
    int c = (id / (LLEN * VV)) & (CC - 1);
    float t = y4[id] * stats[128 + c] + stats[192 + c];
    t = fmaxf(t, 0.f);
    out[id] = fmaxf(t + x[id], 0.f);
}

// --------------------------------------------------------------------------
extern "C" void kernel_launch(void* const* d_in, const int* in_sizes, int n_in,
                              void* d_out, int out_size, void* d_ws, size_t ws_size,
                              hipStream_t stream) {
    const float* x     = (const float*)d_in[0];   // (4,64,1024,25)
    const float* A     = (const float*)d_in[1];   // (3,25,25)
    const float* ei    = (const float*)d_in[2];   // (3,25,25)
    const float* convw = (const float*)d_in[3];   // (192,64,1,1)
    const float* convb = (const float*)d_in[4];   // (192)
    const float* gamma = (const float*)d_in[5];   // (64)
    const float* beta  = (const float*)d_in[6];   // (64)
    float* out = (float*)d_out;

    float* ws    = (float*)d_ws;
    float* mc    = ws + OFF_MC;
    float* xp    = ws + OFF_XP;
    float* Y     = ws + OFF_Y;
    float* y4    = ws + OFF_Y4;
    float* bias  = ws + OFF_BIAS;
    float* stats = ws + OFF_STATS;

    build_mc<<<(KDIM * NDIM + 255) / 256, 256, 0, stream>>>(convw, A, ei, mc);
    build_bias<<<(NDIM + 255) / 256, 256, 0, stream>>>(convb, A, ei, bias, stats);
    pack_x<<<(MDIM * KDIM + 255) / 256, 256, 0, stream>>>(x, xp);

    gemm_wmma<<<dim3(MDIM / 128, NDIM / 64), 256, 0, stream>>>(xp, mc, bias, Y);

    window_sum<<<dim3(10, NN, LLEN / 128), 160, 0, stream>>>(Y, y4, stats);
    finalize_stats<<<1, 64, 0, stream>>>(stats, gamma, beta);
    bn_epilogue<<<(NELEM + 255) / 256, 256, 0, stream>>>(y4, stats, x, out);
}